// CA_event_45638322487767
// MI455X (gfx1250) — compile-verified
//
#include <hip/hip_runtime.h>

// Batched CA_event.forward on MI455X (gfx1250, wave32).
// Pure HBM-streaming elementwise kernel: 40 B of traffic per system at
// ~0.75 flop/byte => bandwidth-bound at 23.3 TB/s (~20 us floor).
// Strategy: 4 systems per thread so every global access is B128, with
// non-temporal (TH=NT) hints since the 470 MB single pass exceeds the
// 192 MB L2 and nothing is re-read.

typedef __attribute__((ext_vector_type(4))) float v4f;

__global__ __launch_bounds__(256) void ca_event_v4(
    const float* __restrict__ x,
    const float* __restrict__ y,
    const float* __restrict__ ex,
    const float* __restrict__ ey,
    const float* __restrict__ wa,      // [B,2] row-major
    const float* __restrict__ target,  // [2], uniform
    float* __restrict__ out,           // [B,4] row-major
    int n4)                            // number of 4-system groups
{
    int t = blockIdx.x * blockDim.x + threadIdx.x;
    if (t >= n4) return;
    const int base = t << 2;  // first system index handled by this thread

    // b128 non-temporal loads (coalesced: lane i covers systems 4i..4i+3)
    v4f xv  = __builtin_nontemporal_load((const v4f*)x  + t);
    v4f yv  = __builtin_nontemporal_load((const v4f*)y  + t);
    v4f exv = __builtin_nontemporal_load((const v4f*)ex + t);
    v4f eyv = __builtin_nontemporal_load((const v4f*)ey + t);
    v4f wa0 = __builtin_nontemporal_load((const v4f*)wa + 2 * t);      // wx0,wy0,wx1,wy1
    v4f wa1 = __builtin_nontemporal_load((const v4f*)wa + 2 * t + 1);  // wx2,wy2,wx3,wy3

    const float tg0 = target[0];  // uniform -> s_load
    const float tg1 = target[1];

    const float SN  = 0.25f;   // S_HALF^2
    const float ISC = 0.1f;    // 1/SCALE
    const float BETA = 0.2f;
    const float KD  = 1.1f;
    const float SCALE = 10.0f;

    const float wax[4] = {wa0.x, wa0.z, wa1.x, wa1.z};
    const float way[4] = {wa0.y, wa0.w, wa1.y, wa1.w};

#pragma unroll
    for (int j = 0; j < 4; ++j) {
        const float xj = xv[j], yj = yv[j];
        const float xs = xj * ISC;
        const float ys = yj * ISC;
        const float xn = xs * xs;
        const float yn = ys * ys;

        // One reciprocal per axis covers both Hill terms:
        //   hx = xn/(sn+xn), ry = sn/(sn+xn);  hy = yn/(sn+yn), rx = sn/(sn+yn)
        const float invx = 1.0f / (SN + xn);
        const float invy = 1.0f / (SN + yn);
        const float hx = xn * invx;
        const float ry = SN * invx;
        const float hy = yn * invy;
        const float rx = SN * invy;

        // control input u = wx*(x+e_x - T) + wy*(y+e_y - T)
        const float u = fmaf(wax[j], (xj + exv[j]) - tg0,
                             way[j] * ((yj + eyv[j]) - tg1));

        // dx = SCALE*((1+u)*hx + 0.2*rx - 1.1*xs)
        const float dx = SCALE * (fmaf(u, hx, hx) + fmaf(BETA, rx, -KD * xs));
        const float dy = SCALE * (hy + fmaf(BETA, ry, -KD * ys));

        v4f o;
        o.x = dx;
        o.y = dy;
        o.z = -dx;
        o.w = -dy;
        // system (base+j) owns out[4*(base+j) .. +3] -> one b128 NT store
        __builtin_nontemporal_store(o, (v4f*)out + (base + j));
    }
}

// Scalar tail (only used if B % 4 != 0; B = 8388608 so normally never launched)
__global__ void ca_event_tail(
    const float* __restrict__ x, const float* __restrict__ y,
    const float* __restrict__ ex, const float* __restrict__ ey,
    const float* __restrict__ wa, const float* __restrict__ target,
    float* __restrict__ out, int start, int n)
{
    int i = start + blockIdx.x * blockDim.x + threadIdx.x;
    if (i >= n) return;
    const float tg0 = target[0], tg1 = target[1];
    const float SN = 0.25f;
    const float xs = x[i] * 0.1f, ys = y[i] * 0.1f;
    const float xn = xs * xs, yn = ys * ys;
    const float invx = 1.0f / (SN + xn), invy = 1.0f / (SN + yn);
    const float hx = xn * invx, ry = SN * invx;
    const float hy = yn * invy, rx = SN * invy;
    const float u = wa[2 * i] * ((x[i] + ex[i]) - tg0) +
                    wa[2 * i + 1] * ((y[i] + ey[i]) - tg1);
    const float dx = 10.0f * (fmaf(u, hx, hx) + fmaf(0.2f, rx, -1.1f * xs));
    const float dy = 10.0f * (hy + fmaf(0.2f, ry, -1.1f * ys));
    out[4 * i + 0] = dx;
    out[4 * i + 1] = dy;
    out[4 * i + 2] = -dx;
    out[4 * i + 3] = -dy;
}

extern "C" void kernel_launch(void* const* d_in, const int* in_sizes, int n_in,
                              void* d_out, int out_size, void* d_ws, size_t ws_size,
                              hipStream_t stream) {
    const float* x      = (const float*)d_in[0];
    const float* y      = (const float*)d_in[1];
    const float* ex     = (const float*)d_in[2];
    const float* ey     = (const float*)d_in[3];
    const float* wa     = (const float*)d_in[4];
    const float* target = (const float*)d_in[5];
    float* out = (float*)d_out;

    const int n  = in_sizes[0];   // B
    const int n4 = n >> 2;        // 4 systems per thread

    if (n4 > 0) {
        const int block = 256;                 // 8 wave32 per block
        const int grid  = (n4 + block - 1) / block;
        ca_event_v4<<<grid, block, 0, stream>>>(x, y, ex, ey, wa, target, out, n4);
    }
    const int tail_start = n4 << 2;
    if (tail_start < n) {
        const int rem = n - tail_start;
        ca_event_tail<<<(rem + 255) / 256, 256, 0, stream>>>(
            x, y, ex, ey, wa, target, out, tail_start, n);
    }
}